// GraphEncoderTransformer_6536940224756
// MI455X (gfx1250) — compile-verified
//
#include <hip/hip_runtime.h>
#include <hip/hip_bf16.h>
#include <float.h>

typedef __attribute__((ext_vector_type(2))) float v2f;
typedef __attribute__((ext_vector_type(8))) float v8f;

#define WAVE 32

// ---------------------------------------------------------------------------
// Generic fp32 GEMM with bias: OUT[N x M] = X[N x K] @ W[K x M] + b[M]
// Uses V_WMMA_F32_16X16X4_F32. One block = one 16-row strip, 8 waves cover
// column tiles. Nrows % 16 == 0, Mcols % 16 == 0, Kdim % 4 == 0, Kdim <= 128.
// EXEC is all-ones everywhere a WMMA executes (no divergence).
// ---------------------------------------------------------------------------
__global__ __launch_bounds__(256)
void gemm_bias_wmma(const float* __restrict__ X, const float* __restrict__ W,
                    const float* __restrict__ bias, float* __restrict__ OUT,
                    int Kdim, int Mcols) {
  __shared__ float Xs[16 * 128];
  const int m0 = blockIdx.x * 16;
  const int tid = threadIdx.x;

  // Cooperative load of the 16 x Kdim A-strip into LDS.
  for (int i = tid; i < 16 * Kdim; i += 256) {
    int r = i / Kdim, c = i - r * Kdim;
    Xs[r * 128 + c] = X[(size_t)(m0 + r) * Kdim + c];
  }
  __syncthreads();

  const int wave = tid >> 5;
  const int lane = tid & 31;
  const int l15  = lane & 15;        // row within A tile / col within B,C tiles
  const int kk   = (lane >> 4) * 2;  // K sub-offset per A/B fragment layout
  const int nTiles = Mcols >> 4;

  for (int nt = wave; nt < nTiles; nt += 8) {
    const int n0 = nt << 4;
    v8f c = {};
    for (int k = 0; k < Kdim; k += 4) {
      v2f a, b;
      // A 16x4 f32 fragment: lanes0-15 K=k..k+1, lanes16-31 K=k+2..k+3
      a.x = Xs[l15 * 128 + k + kk];
      a.y = Xs[l15 * 128 + k + kk + 1];
      // B 4x16 f32 fragment: VGPR0 K=kk, VGPR1 K=kk+1; N = n0 + l15
      b.x = W[(size_t)(k + kk) * Mcols + n0 + l15];
      b.y = W[(size_t)(k + kk + 1) * Mcols + n0 + l15];
      c = __builtin_amdgcn_wmma_f32_16x16x4_f32(false, a, false, b,
                                                (short)0, c, false, false);
    }
    // C/D layout: VGPR r -> M = r + 8*(lane>>4), N = n0 + l15
    const int col   = n0 + l15;
    const float bv  = bias[col];
    const int rbase = m0 + ((lane >> 4) << 3);
#pragma unroll
    for (int r = 0; r < 8; ++r)
      OUT[(size_t)(rbase + r) * Mcols + col] = c[r] + bv;
  }
}

// ---------------------------------------------------------------------------
// Utility fills (workspace is poisoned 0xAA before timing; re-init each call)
// ---------------------------------------------------------------------------
__global__ void fill_u32(unsigned* __restrict__ p, unsigned v, size_t n) {
  size_t i = (size_t)blockIdx.x * blockDim.x + threadIdx.x;
  if (i < n) p[i] = v;
}

// Sign-aware float atomic-max via integer atomics (works for mixed signs,
// target initialized to -FLT_MAX).
__device__ __forceinline__ void atomicMaxF(float* addr, float val) {
  if (val >= 0.0f) atomicMax((int*)addr, __float_as_int(val));
  else             atomicMin((unsigned int*)addr, __float_as_uint(val));
}

// ---------------------------------------------------------------------------
// Edge kernels. One wave32 per edge. HEADS=4, HEAD_DIM=32 -> flat dim 128.
// ---------------------------------------------------------------------------
__global__ __launch_bounds__(256)
void edge_logits_kernel(const int* __restrict__ src, const int* __restrict__ dst,
                        const float* __restrict__ Q, const float* __restrict__ Kf,
                        float* __restrict__ logits, float* __restrict__ mbuf, int E) {
  int e = blockIdx.x * 8 + (threadIdx.x >> 5);
  if (e >= E) return;
  const int lane = threadIdx.x & 31;
  const int s = src[e], d = dst[e];
  const float* q = Q  + (size_t)d * 128;   // query at target
  const float* k = Kf + (size_t)s * 128;   // key at source
  float l0 = q[lane]      * k[lane];
  float l1 = q[32 + lane] * k[32 + lane];
  float l2 = q[64 + lane] * k[64 + lane];
  float l3 = q[96 + lane] * k[96 + lane];
#pragma unroll
  for (int off = 16; off > 0; off >>= 1) {
    l0 += __shfl_xor(l0, off, WAVE);
    l1 += __shfl_xor(l1, off, WAVE);
    l2 += __shfl_xor(l2, off, WAVE);
    l3 += __shfl_xor(l3, off, WAVE);
  }
  if (lane < 4) {
    float L = (lane == 0) ? l0 : (lane == 1) ? l1 : (lane == 2) ? l2 : l3;
    L *= 0.17677669529663687f;   // 1/sqrt(32)
    logits[(size_t)e * 4 + lane] = L;
    atomicMaxF(&mbuf[(size_t)d * 4 + lane], L);
  }
}

__global__ __launch_bounds__(256)
void edge_p_kernel(const int* __restrict__ dst, float* __restrict__ logits,
                   const float* __restrict__ mbuf, float* __restrict__ denom, int E) {
  size_t i = (size_t)blockIdx.x * blockDim.x + threadIdx.x;  // over E*4
  if (i >= (size_t)E * 4) return;
  int e = (int)(i >> 2), h = (int)(i & 3);
  int d = dst[e];
  float p = __expf(logits[i] - mbuf[(size_t)d * 4 + h]);
  logits[i] = p;
  atomicAdd(&denom[(size_t)d * 4 + h], p);
}

// Fuses attention-weighted message with head-mean: 32 atomics/edge (not 128).
__global__ __launch_bounds__(256)
void edge_agg_kernel(const int* __restrict__ src, const int* __restrict__ dst,
                     const float* __restrict__ V, const float* __restrict__ logits,
                     const float* __restrict__ denom, float* __restrict__ agg, int E) {
  int e = blockIdx.x * 8 + (threadIdx.x >> 5);
  if (e >= E) return;
  const int lane = threadIdx.x & 31;
  const int s = src[e], d = dst[e];
  const float* lp = logits + (size_t)e * 4;
  const float* dp = denom  + (size_t)d * 4;
  const float a0 = lp[0] / fmaxf(dp[0], 1e-16f);
  const float a1 = lp[1] / fmaxf(dp[1], 1e-16f);
  const float a2 = lp[2] / fmaxf(dp[2], 1e-16f);
  const float a3 = lp[3] / fmaxf(dp[3], 1e-16f);
  const float* v = V + (size_t)s * 128;
  const float acc = 0.25f * (a0 * v[lane] + a1 * v[32 + lane] +
                             a2 * v[64 + lane] + a3 * v[96 + lane]);
  atomicAdd(&agg[(size_t)d * 32 + lane], acc);
}

// h = relu(agg + skip)   (skip already holds x @ Ws + bs; in-place on skip)
__global__ void finalize_relu_kernel(const float* __restrict__ agg,
                                     float* __restrict__ h, size_t n) {
  size_t i = (size_t)blockIdx.x * blockDim.x + threadIdx.x;
  if (i < n) h[i] = fmaxf(agg[i] + h[i], 0.0f);
}

// ---------------------------------------------------------------------------
// Global mean pool over 512 graphs (H=32). One wave per node.
// ---------------------------------------------------------------------------
__global__ __launch_bounds__(256)
void pool_accum_kernel(const float* __restrict__ h, const int* __restrict__ batch,
                       float* __restrict__ pool, float* __restrict__ cnt, int N) {
  int n = blockIdx.x * 8 + (threadIdx.x >> 5);
  if (n >= N) return;
  const int lane = threadIdx.x & 31;
  const int g = batch[n];
  atomicAdd(&pool[(size_t)g * 32 + lane], h[(size_t)n * 32 + lane]);
  if (lane == 0) atomicAdd(&cnt[g], 1.0f);
}

__global__ void pool_div_kernel(const float* __restrict__ pool,
                                const float* __restrict__ cnt,
                                float* __restrict__ out, int n) {
  int i = blockIdx.x * blockDim.x + threadIdx.x;
  if (i < n) out[i] = pool[i] / fmaxf(cnt[i >> 5], 1.0f);
}

// ---------------------------------------------------------------------------
static inline int cdiv(long long a, long long b) { return (int)((a + b - 1) / b); }

static void run_tconv(const float* xin, int Kdim,
                      const float* Wq, const float* bq, const float* Wk, const float* bk,
                      const float* Wv, const float* bv, const float* Ws, const float* bs,
                      float* Qb, float* Kb, float* Vb, float* hout,
                      const int* src, const int* dst,
                      float* lg, float* mb, float* dn, float* ag,
                      int N, int E, hipStream_t stream) {
  const int rowBlocks = N / 16;  // N = 100000 -> 6250 exact
  gemm_bias_wmma<<<rowBlocks, 256, 0, stream>>>(xin, Wq, bq, Qb, Kdim, 128);
  gemm_bias_wmma<<<rowBlocks, 256, 0, stream>>>(xin, Wk, bk, Kb, Kdim, 128);
  gemm_bias_wmma<<<rowBlocks, 256, 0, stream>>>(xin, Wv, bv, Vb, Kdim, 128);
  gemm_bias_wmma<<<rowBlocks, 256, 0, stream>>>(xin, Ws, bs, hout, Kdim, 32);

  // init: mb = -FLT_MAX (bits 0xFF7FFFFF); dn,ag = 0 (contiguous -> one fill)
  fill_u32<<<cdiv((size_t)N * 4, 256), 256, 0, stream>>>((unsigned*)mb, 0xFF7FFFFFu, (size_t)N * 4);
  fill_u32<<<cdiv((size_t)N * 36, 256), 256, 0, stream>>>((unsigned*)dn, 0u, (size_t)N * 36);

  edge_logits_kernel<<<cdiv(E, 8), 256, 0, stream>>>(src, dst, Qb, Kb, lg, mb, E);
  edge_p_kernel<<<cdiv((size_t)E * 4, 256), 256, 0, stream>>>(dst, lg, mb, dn, E);
  edge_agg_kernel<<<cdiv(E, 8), 256, 0, stream>>>(src, dst, Vb, lg, dn, ag, E);
  finalize_relu_kernel<<<cdiv((size_t)N * 32, 256), 256, 0, stream>>>(ag, hout, (size_t)N * 32);
}

extern "C" void kernel_launch(void* const* d_in, const int* in_sizes, int n_in,
                              void* d_out, int out_size, void* d_ws, size_t ws_size,
                              hipStream_t stream) {
  const float* x     = (const float*)d_in[0];
  const int*   ei    = (const int*)d_in[1];
  const int*   batch = (const int*)d_in[2];
  const float *Wq1 = (const float*)d_in[3],  *bq1 = (const float*)d_in[4];
  const float *Wk1 = (const float*)d_in[5],  *bk1 = (const float*)d_in[6];
  const float *Wv1 = (const float*)d_in[7],  *bv1 = (const float*)d_in[8];
  const float *Ws1 = (const float*)d_in[9],  *bs1 = (const float*)d_in[10];
  const float *Wq2 = (const float*)d_in[11], *bq2 = (const float*)d_in[12];
  const float *Wk2 = (const float*)d_in[13], *bk2 = (const float*)d_in[14];
  const float *Wv2 = (const float*)d_in[15], *bv2 = (const float*)d_in[16];
  const float *Ws2 = (const float*)d_in[17], *bs2 = (const float*)d_in[18];

  const int N = in_sizes[2];          // 100000
  const int E = in_sizes[1] / 2;      // 1600000
  const int* src = ei;
  const int* dst = ei + E;

  // Workspace layout (floats)
  float* ws = (float*)d_ws;
  size_t off = 0;
  float* Qb = ws + off; off += (size_t)N * 128;
  float* Kb = ws + off; off += (size_t)N * 128;
  float* Vb = ws + off; off += (size_t)N * 128;
  float* lg = ws + off; off += (size_t)E * 4;      // per-edge logits -> p
  float* mb = ws + off; off += (size_t)N * 4;      // segment max
  float* dn = ws + off; off += (size_t)N * 4;      // segment sum   (dn,ag contiguous)
  float* ag = ws + off; off += (size_t)N * 32;     // head-mean aggregate
  float* h1 = ws + off; off += (size_t)N * 32;
  float* h2 = ws + off; off += (size_t)N * 32;
  float* pool = ws + off; off += (size_t)512 * 32; // pool,cnt contiguous
  float* cnt  = ws + off; off += 512;

  // Layer 1: x[N,128] -> h1[N,32]
  run_tconv(x, 128, Wq1, bq1, Wk1, bk1, Wv1, bv1, Ws1, bs1,
            Qb, Kb, Vb, h1, src, dst, lg, mb, dn, ag, N, E, stream);
  // Layer 2: h1[N,32] -> h2[N,32]
  run_tconv(h1, 32, Wq2, bq2, Wk2, bk2, Wv2, bv2, Ws2, bs2,
            Qb, Kb, Vb, h2, src, dst, lg, mb, dn, ag, N, E, stream);

  // Global mean pool over 512 graphs
  fill_u32<<<cdiv(512 * 32 + 512, 256), 256, 0, stream>>>((unsigned*)pool, 0u, 512 * 32 + 512);
  pool_accum_kernel<<<cdiv(N, 8), 256, 0, stream>>>(h2, batch, pool, cnt, N);
  pool_div_kernel<<<cdiv(512 * 32, 256), 256, 0, stream>>>(pool, cnt, (float*)d_out, 512 * 32);
}